// TgPredictorGNN_v2_18262200942604
// MI455X (gfx1250) — compile-verified
//
#include <hip/hip_runtime.h>
#include <hip/hip_bf16.h>

#define N_NODES    100000
#define N_EDGES    1600000
#define NODE_DIM   64
#define HIDDEN     64
#define NUM_GRAPHS 1024

typedef __attribute__((ext_vector_type(2))) float v2f;
typedef __attribute__((ext_vector_type(8))) float v8f;

// ---------------------------------------------------------------------------
// Edge weight: ew[e] = softplus(edge_attr[e,1:4] . w) * edge_attr[e,4]
// ---------------------------------------------------------------------------
__global__ __launch_bounds__(256) void edge_w_kernel(
    const float* __restrict__ ea, const float* __restrict__ w,
    float* __restrict__ ew, int ne)
{
    int e = blockIdx.x * 256 + threadIdx.x;
    if (e >= ne) return;
    const float* a = ea + (size_t)e * 5;
    float z  = a[1] * w[0] + a[2] * w[1] + a[3] * w[2];
    float sp = (z > 0.0f) ? (z + log1pf(expf(-z))) : log1pf(expf(z));
    ew[e] = sp * a[4];
}

// deg[i] = 1.0 (self-loop weight), then atomically accumulate edge weights.
__global__ __launch_bounds__(256) void deg_init_kernel(float* __restrict__ deg, int n)
{
    int i = blockIdx.x * 256 + threadIdx.x;
    if (i < n) deg[i] = 1.0f;
}

__global__ __launch_bounds__(256) void deg_acc_kernel(
    const int* __restrict__ dst, const float* __restrict__ ew,
    float* __restrict__ deg, int ne)
{
    int e = blockIdx.x * 256 + threadIdx.x;
    if (e >= ne) return;
    atomicAdd(&deg[dst[e]], ew[e]);
}

// deg -> dinv in place
__global__ __launch_bounds__(256) void dinv_kernel(float* __restrict__ deg, int n)
{
    int i = blockIdx.x * 256 + threadIdx.x;
    if (i >= n) return;
    float d = deg[i];
    deg[i] = (d > 0.0f) ? rsqrtf(fmaxf(d, 1e-30f)) : 0.0f;
}

// norm[e] = dinv[src] * ew[e] * dinv[dst]
__global__ __launch_bounds__(256) void norm_kernel(
    const int* __restrict__ src, const int* __restrict__ dst,
    const float* __restrict__ ew, const float* __restrict__ dinv,
    float* __restrict__ nrm, int ne)
{
    int e = blockIdx.x * 256 + threadIdx.x;
    if (e >= ne) return;
    nrm[e] = dinv[src[e]] * ew[e] * dinv[dst[e]];
}

// ---------------------------------------------------------------------------
// GEMM: C[M,64] = A[M,64] @ W[64,64] via V_WMMA_F32_16X16X4_F32.
// One wave computes a 16x64 tile; 8 waves/block.
// W is staged in LDS K-pair-interleaved: element (k,n) at
// Wl[(k>>1)*128 + n*2 + (k&1)], so each lane's B fragment {W[k][n],W[k+1][n]}
// is one contiguous ds_load_b64 into an even-aligned VGPR pair (no shuffles).
// ---------------------------------------------------------------------------
__global__ __launch_bounds__(256) void gemm_wmma_kernel(
    const float* __restrict__ A, const float* __restrict__ W,
    float* __restrict__ C, int M)
{
    __shared__ float Wl[64 * 64];
    for (int i = threadIdx.x; i < 64 * 64; i += 256) {
        int k = i >> 6;
        int n = i & 63;
        Wl[(k >> 1) * 128 + n * 2 + (k & 1)] = W[i];
    }
    __syncthreads();

    const int wave = threadIdx.x >> 5;
    const int lane = threadIdx.x & 31;
    const int hl   = lane >> 4;      // half: 0 = lanes 0-15, 1 = lanes 16-31
    const int m    = lane & 15;

    const int numTiles = M >> 4;     // M is a multiple of 16
    const int rowTile  = blockIdx.x * 8 + wave;
    if (rowTile >= numTiles) return; // wave-uniform guard (EXEC stays all-1s)
    const int row0 = rowTile << 4;

    v8f acc0 = {}, acc1 = {}, acc2 = {}, acc3 = {};
    const float* arow = A + (size_t)(row0 + m) * 64;

    #pragma unroll
    for (int k0 = 0; k0 < 64; k0 += 4) {
        // A fragment (16x4 f32): lanes 0-15 hold K=k0,k0+1; lanes 16-31 K=k0+2,k0+3
        v2f a = *(const v2f*)(arow + k0 + 2 * hl);
        // B fragments: one b64 LDS read per N-tile
        const v2f* bp = (const v2f*)&Wl[((k0 + 2 * hl) >> 1) * 128];
        v2f b0 = bp[m +  0];
        v2f b1 = bp[m + 16];
        v2f b2 = bp[m + 32];
        v2f b3 = bp[m + 48];
        acc0 = __builtin_amdgcn_wmma_f32_16x16x4_f32(false, a, false, b0, (short)0, acc0, false, false);
        acc1 = __builtin_amdgcn_wmma_f32_16x16x4_f32(false, a, false, b1, (short)0, acc1, false, false);
        acc2 = __builtin_amdgcn_wmma_f32_16x16x4_f32(false, a, false, b2, (short)0, acc2, false, false);
        acc3 = __builtin_amdgcn_wmma_f32_16x16x4_f32(false, a, false, b3, (short)0, acc3, false, false);
    }

    // C/D layout: VGPR r -> rows (row0 + r) for lanes 0-15, (row0 + r + 8) for lanes 16-31
    float* crow = C + (size_t)(row0 + 8 * hl) * 64 + m;
    #pragma unroll
    for (int r = 0; r < 8; ++r) {
        crow[r * 64 +  0] = acc0[r];
        crow[r * 64 + 16] = acc1[r];
        crow[r * 64 + 32] = acc2[r];
        crow[r * 64 + 48] = acc3[r];
    }
}

// ---------------------------------------------------------------------------
// Aggregation: out[i] = h[i]*dinv[i]^2 (self-loop), then edge scatter-adds,
// then out = relu(out + bias).
// ---------------------------------------------------------------------------
__global__ __launch_bounds__(256) void agg_init_kernel(
    const float* __restrict__ h, const float* __restrict__ dinv,
    float* __restrict__ out, int total)
{
    int idx = blockIdx.x * 256 + threadIdx.x;
    if (idx >= total) return;
    int node = idx >> 6;
    float di = dinv[node];
    out[idx] = h[idx] * di * di;
}

__global__ __launch_bounds__(256) void agg_edges_kernel(
    const float* __restrict__ h,
    const int* __restrict__ src, const int* __restrict__ dst,
    const float* __restrict__ nrm, float* __restrict__ out, int ne)
{
    int e    = blockIdx.x * 8 + (threadIdx.x >> 5);
    int lane = threadIdx.x & 31;
    if (e >= ne) return;
    int   s  = src[e];
    int   d  = dst[e];
    float nm = nrm[e];
    const float2 hv = *(const float2*)(h + (size_t)s * 64 + lane * 2);
    float* o = out + (size_t)d * 64 + lane * 2;
    atomicAdd(o + 0, hv.x * nm);
    atomicAdd(o + 1, hv.y * nm);
}

__global__ __launch_bounds__(256) void relu_bias_kernel(
    float* __restrict__ out, const float* __restrict__ bias, int total)
{
    int idx = blockIdx.x * 256 + threadIdx.x;
    if (idx >= total) return;
    out[idx] = fmaxf(out[idx] + bias[idx & 63], 0.0f);
}

// ---------------------------------------------------------------------------
// Mean pooling over graphs + counts
// ---------------------------------------------------------------------------
__global__ __launch_bounds__(256) void pool_zero_kernel(
    float* __restrict__ pooled, float* __restrict__ counts, int np, int ng)
{
    int i = blockIdx.x * 256 + threadIdx.x;
    if (i < np) pooled[i] = 0.0f;
    if (i < ng) counts[i] = 0.0f;
}

__global__ __launch_bounds__(256) void pool_kernel(
    const float* __restrict__ h, const int* __restrict__ batch,
    float* __restrict__ pooled, float* __restrict__ counts, int total)
{
    int idx = blockIdx.x * 256 + threadIdx.x;
    if (idx >= total) return;
    int node = idx >> 6;
    int c    = idx & 63;
    int g    = batch[node];
    atomicAdd(&pooled[g * 64 + c], h[idx]);
    if (c == 0) atomicAdd(&counts[g], 1.0f);
}

// ---------------------------------------------------------------------------
// MLP head: out[g] = (relu(mean_pool @ fc1 + b1)) @ fc2 + b2
// One wave (32 threads) per graph; lane j owns fc1 unit j.
// ---------------------------------------------------------------------------
__global__ __launch_bounds__(32) void mlp_kernel(
    const float* __restrict__ pooled, const float* __restrict__ counts,
    const float* __restrict__ f1w, const float* __restrict__ f1b,
    const float* __restrict__ f2w, const float* __restrict__ f2b,
    float* __restrict__ out)
{
    int g = blockIdx.x;
    int j = threadIdx.x;
    float inv = 1.0f / fmaxf(counts[g], 1.0f);
    const float* p = pooled + g * 64;
    float s = f1b[j];
    #pragma unroll
    for (int k = 0; k < 64; ++k) s += p[k] * inv * f1w[k * 32 + j];
    s = fmaxf(s, 0.0f);
    float v = s * f2w[j];
    #pragma unroll
    for (int off = 16; off; off >>= 1) v += __shfl_xor(v, off, 32);
    if (j == 0) out[g] = v + f2b[0];
}

// ---------------------------------------------------------------------------
extern "C" void kernel_launch(void* const* d_in, const int* in_sizes, int n_in,
                              void* d_out, int out_size, void* d_ws, size_t ws_size,
                              hipStream_t stream)
{
    const float* x   = (const float*)d_in[0];
    const float* ea  = (const float*)d_in[1];
    const float* w   = (const float*)d_in[2];
    const float* W1  = (const float*)d_in[3];
    const float* b1  = (const float*)d_in[4];
    const float* W2  = (const float*)d_in[5];
    const float* b2  = (const float*)d_in[6];
    const float* f1w = (const float*)d_in[7];
    const float* f1b = (const float*)d_in[8];
    const float* f2w = (const float*)d_in[9];
    const float* f2b = (const float*)d_in[10];
    const int* eidx  = (const int*)d_in[11];
    const int* batch = (const int*)d_in[12];
    const int* src = eidx;
    const int* dst = eidx + N_EDGES;

    float* ws     = (float*)d_ws;
    float* ew     = ws;  ws += N_EDGES;
    float* nrm    = ws;  ws += N_EDGES;
    float* dinv   = ws;  ws += N_NODES;
    float* h      = ws;  ws += (size_t)N_NODES * 64;
    float* hb     = ws;  ws += (size_t)N_NODES * 64;
    float* pooled = ws;  ws += NUM_GRAPHS * 64;
    float* counts = ws;  ws += NUM_GRAPHS;

    const int totalNC   = N_NODES * 64;
    const int gridEdges = (N_EDGES + 255) / 256;          // 6250
    const int gridNodes = (N_NODES + 255) / 256;          // 391
    const int gridNC    = (totalNC + 255) / 256;          // 25000
    const int gridAgg   = (N_EDGES + 7) / 8;              // 200000
    const int gridGemm  = ((N_NODES / 16) + 7) / 8;       // 782

    // edge weights + symmetric normalization
    edge_w_kernel  <<<gridEdges, 256, 0, stream>>>(ea, w, ew, N_EDGES);
    deg_init_kernel<<<gridNodes, 256, 0, stream>>>(dinv, N_NODES);
    deg_acc_kernel <<<gridEdges, 256, 0, stream>>>(dst, ew, dinv, N_EDGES);
    dinv_kernel    <<<gridNodes, 256, 0, stream>>>(dinv, N_NODES);
    norm_kernel    <<<gridEdges, 256, 0, stream>>>(src, dst, ew, dinv, nrm, N_EDGES);

    // conv1: h = x @ W1 ; hb = aggregate ; relu(+b1)
    gemm_wmma_kernel<<<gridGemm, 256, 0, stream>>>(x, W1, h, N_NODES);
    agg_init_kernel <<<gridNC,   256, 0, stream>>>(h, dinv, hb, totalNC);
    agg_edges_kernel<<<gridAgg,  256, 0, stream>>>(h, src, dst, nrm, hb, N_EDGES);
    relu_bias_kernel<<<gridNC,   256, 0, stream>>>(hb, b1, totalNC);

    // conv2: h = hb @ W2 ; hb = aggregate ; relu(+b2)
    gemm_wmma_kernel<<<gridGemm, 256, 0, stream>>>(hb, W2, h, N_NODES);
    agg_init_kernel <<<gridNC,   256, 0, stream>>>(h, dinv, hb, totalNC);
    agg_edges_kernel<<<gridAgg,  256, 0, stream>>>(h, src, dst, nrm, hb, N_EDGES);
    relu_bias_kernel<<<gridNC,   256, 0, stream>>>(hb, b2, totalNC);

    // mean pool + MLP head
    pool_zero_kernel<<<(NUM_GRAPHS * 64 + 255) / 256, 256, 0, stream>>>(
        pooled, counts, NUM_GRAPHS * 64, NUM_GRAPHS);
    pool_kernel<<<gridNC, 256, 0, stream>>>(hb, batch, pooled, counts, totalNC);
    mlp_kernel <<<NUM_GRAPHS, 32, 0, stream>>>(pooled, counts, f1w, f1b, f2w, f2b,
                                               (float*)d_out);
}